// resize_module_58325655879844
// MI455X (gfx1250) — compile-verified
//
#include <hip/hip_runtime.h>

typedef __attribute__((ext_vector_type(2))) float v2f;
typedef __attribute__((ext_vector_type(8))) float v8f;

#define H1 32
#define W1 32
#define H2 16
#define W2 16
#define NSLAB (128 * 512)          // N*C independent 32x32 -> 16x16 slabs
#define TSTRIDE 34                 // LDS row stride (dwords): conflict-free + 8B aligned
#define WAVES_PER_BLOCK 8
#define SLABS_PER_WAVE 8

// antiderivative of max(0, 1-|t|), clamped to support (matches jnp.clip + piecewise G)
__device__ __forceinline__ float triG(float t) {
    t = fminf(1.0f, fmaxf(-1.0f, t));
    float a = t + 1.0f;
    return (t <= 0.0f) ? 0.5f * a * a : 0.5f + t - 0.5f * t * t;
}

// W[i,p] * (16/31): triangle-kernel integral over bin i at grid point p,
// pre-scaled so Wy*X*Wx^T directly includes the 1/(bin_h*bin_w) factor.
__device__ __forceinline__ float binw(int i, int p) {
    const float bin = 31.0f / 16.0f;
    float e0 = bin * (float)i;
    float e1 = bin * (float)(i + 1);
    float fp = (float)p;
    return (triG(e1 - fp) - triG(e0 - fp)) * (16.0f / 31.0f);
}

__device__ __forceinline__ v8f wmma_f32(v2f a, v2f b, v8f c) {
    // (neg_a, A, neg_b, B, c_mod, C, reuse_a, reuse_b)
    return __builtin_amdgcn_wmma_f32_16x16x4_f32(false, a, false, b, (short)0, c,
                                                 false, false);
}

__global__ __launch_bounds__(32 * WAVES_PER_BLOCK)
void prroi_wmma_kernel(const float* __restrict__ x, float* __restrict__ out) {
    __shared__ float lds[WAVES_PER_BLOCK * H2 * TSTRIDE];

    const int lane  = threadIdx.x & 31;
    const int lhalf = lane & 15;        // M / N / column-in-halfwave index
    const int hi    = lane >> 4;        // 0: lanes 0-15, 1: lanes 16-31
    const int wave  = threadIdx.x >> 5;
    float* tb = lds + wave * (H2 * TSTRIDE);   // wave-private 16 x 32 (stride 34) tile

    // Shared weight registers: serve as stage-1 A-tiles (Wy[:,4k:4k+4], 16x4 layout)
    // AND stage-2 B-tiles (Wx^T[4k:4k+4,:], 4x16 layout) — per-lane values coincide.
    v2f wgt[8];
#pragma unroll
    for (int k = 0; k < 8; ++k) {
        int p = 4 * k + 2 * hi;
        wgt[k].x = binw(lhalf, p);
        wgt[k].y = binw(lhalf, p + 1);
    }

    const int gw = blockIdx.x * WAVES_PER_BLOCK + wave;

#pragma unroll 1
    for (int s = 0; s < SLABS_PER_WAVE; ++s) {
        const int slab = gw * SLABS_PER_WAVE + s;
        const float* xs = x + (size_t)slab * (H1 * W1);

        // ---- stage 1: T[16x32] = Wy(16x32) @ X(32x32), as two 16x16 N-tiles, K=32
        v8f T0 = {0.f, 0.f, 0.f, 0.f, 0.f, 0.f, 0.f, 0.f};
        v8f T1 = {0.f, 0.f, 0.f, 0.f, 0.f, 0.f, 0.f, 0.f};
#pragma unroll
        for (int k = 0; k < 8; ++k) {
            const int r0 = 4 * k + 2 * hi;     // X rows r0, r0+1 feed this lane's B regs
            v2f b0, b1;                        // 4x16 B-layout slices of X
            b0.x = __builtin_nontemporal_load(xs + r0 * W1 + lhalf);
            b0.y = __builtin_nontemporal_load(xs + (r0 + 1) * W1 + lhalf);
            b1.x = __builtin_nontemporal_load(xs + r0 * W1 + 16 + lhalf);
            b1.y = __builtin_nontemporal_load(xs + (r0 + 1) * W1 + 16 + lhalf);
            T0 = wmma_f32(wgt[k], b0, T0);
            T1 = wmma_f32(wgt[k], b1, T1);
        }

        // ---- transpose T (D-layout -> A-layout) through wave-private LDS
#pragma unroll
        for (int r = 0; r < 8; ++r) {
            const int row = r + 8 * hi;                  // D-layout row of VGPR r
            tb[row * TSTRIDE + lhalf]      = T0[r];
            tb[row * TSTRIDE + 16 + lhalf] = T1[r];
        }
        __asm__ volatile("s_wait_dscnt 0x0" ::: "memory");

        // ---- stage 2: Out[16x16] = T(16x32) @ Wx^T(32x16), K=32
        v8f O = {0.f, 0.f, 0.f, 0.f, 0.f, 0.f, 0.f, 0.f};
#pragma unroll
        for (int k = 0; k < 8; ++k) {
            const int col = 4 * k + 2 * hi;              // A-layout: T[lhalf, col..col+1]
            v2f a = *(const v2f*)(tb + lhalf * TSTRIDE + col);  // ds_load_b64
            O = wmma_f32(a, wgt[k], O);
        }
        __asm__ volatile("s_wait_dscnt 0x0" ::: "memory");  // guard LDS reuse next iter

        // ---- store D-layout straight to row-major output
        float* os = out + (size_t)slab * (H2 * W2);
#pragma unroll
        for (int r = 0; r < 8; ++r) {
            __builtin_nontemporal_store(O[r], os + (r + 8 * hi) * W2 + lhalf);
        }
    }
}

extern "C" void kernel_launch(void* const* d_in, const int* in_sizes, int n_in,
                              void* d_out, int out_size, void* d_ws, size_t ws_size,
                              hipStream_t stream) {
    (void)in_sizes; (void)n_in; (void)out_size; (void)d_ws; (void)ws_size;
    const float* x = (const float*)d_in[0];
    float* out = (float*)d_out;
    // 65536 slabs / (8 waves * 8 slabs per wave) = 1024 blocks of 256 threads
    dim3 grid(NSLAB / (WAVES_PER_BLOCK * SLABS_PER_WAVE));
    dim3 block(32 * WAVES_PER_BLOCK);
    prroi_wmma_kernel<<<grid, block, 0, stream>>>(x, out);
}